// EigenActivation_89876485636498
// MI455X (gfx1250) — compile-verified
//
#include <hip/hip_runtime.h>
#include <math.h>

// WMMA fragment vector types (wave32, gfx1250)
typedef __attribute__((ext_vector_type(2))) float v2f;   // 16x4 f32 A / 4x16 f32 B fragment
typedef __attribute__((ext_vector_type(8))) float v8f;   // 16x16 f32 C/D fragment

#define DIM 64
#define LDA 65          // LDS row stride (65 mod 64 banks -> conflict-free column walks)
#define NPAIR 32        // disjoint Jacobi pairs per round
#define NSWEEP 8        // cyclic Jacobi sweeps (64x64 fp32 converges in ~6-10)

// Round-robin (circle-method) pairing: round r in [0,63), pair k in [0,32)
__device__ __forceinline__ void pair_pq(int r, int k, int& p, int& q) {
    if (k == 0) { p = 63; q = r; }
    else {
        p = (r + k) % 63;
        q = (r - k + 63) % 63;
    }
}

__global__ __launch_bounds__(256)
void eigact_reeig_kernel(const float* __restrict__ X, float* __restrict__ Out) {
    __shared__ float sA[DIM * LDA];   // working matrix, later M = Q*diag(ev)
    __shared__ float sQ[DIM * LDA];   // accumulated eigenvectors (columns)
    __shared__ float sC[NPAIR];
    __shared__ float sS[NPAIR];
    __shared__ float sEv[DIM];

    const int t = threadIdx.x;
    const int b = blockIdx.x;
    const float* __restrict__ Xb = X + (size_t)b * DIM * DIM;
    float* __restrict__ Ob = Out + (size_t)b * DIM * DIM;

    // ---- Load A (vectorized b128 reads), init Q = I ----
    for (int id = t; id < DIM * DIM / 4; id += 256) {
        const float4 v = ((const float4*)Xb)[id];
        const int row = id >> 4;           // 16 float4 per row
        const int col = (id & 15) * 4;
        sA[row * LDA + col + 0] = v.x;
        sA[row * LDA + col + 1] = v.y;
        sA[row * LDA + col + 2] = v.z;
        sA[row * LDA + col + 3] = v.w;
    }
    for (int id = t; id < DIM * DIM; id += 256) {
        const int row = id >> 6, col = id & 63;
        sQ[row * LDA + col] = (row == col) ? 1.0f : 0.0f;
    }
    __syncthreads();

    // ---- Parallel cyclic Jacobi: A <- J^T A J, Q <- Q J ----
    for (int sweep = 0; sweep < NSWEEP; ++sweep) {
        for (int r = 0; r < 63; ++r) {
            // 1) rotation angles for the 32 disjoint pairs
            if (t < NPAIR) {
                int p, q; pair_pq(r, t, p, q);
                const float app = sA[p * LDA + p];
                const float aqq = sA[q * LDA + q];
                const float apq = sA[p * LDA + q];
                float c = 1.0f, s = 0.0f;
                if (fabsf(apq) > 1e-12f) {
                    const float tau = (aqq - app) / (2.0f * apq);
                    const float tt  = copysignf(1.0f / (fabsf(tau) + sqrtf(1.0f + tau * tau)), tau);
                    c = rsqrtf(1.0f + tt * tt);
                    s = tt * c;
                }
                sC[t] = c; sS[t] = s;
            }
            __syncthreads();
            // 2) column update: A <- A*J and Q <- Q*J  (columns p,q disjoint across pairs)
            for (int id = t; id < DIM * NPAIR; id += 256) {
                const int k = id >> 6, i = id & 63;
                int p, q; pair_pq(r, k, p, q);
                const float c = sC[k], s = sS[k];
                const float ap = sA[i * LDA + p], aq = sA[i * LDA + q];
                sA[i * LDA + p] = c * ap - s * aq;
                sA[i * LDA + q] = s * ap + c * aq;
                const float qp = sQ[i * LDA + p], qq = sQ[i * LDA + q];
                sQ[i * LDA + p] = c * qp - s * qq;
                sQ[i * LDA + q] = s * qp + c * qq;
            }
            __syncthreads();
            // 3) row update: A <- J^T * A  (rows p,q disjoint across pairs)
            for (int id = t; id < DIM * NPAIR; id += 256) {
                const int k = id >> 6, j = id & 63;
                int p, q; pair_pq(r, k, p, q);
                const float c = sC[k], s = sS[k];
                const float ap = sA[p * LDA + j], aq = sA[q * LDA + j];
                sA[p * LDA + j] = c * ap - s * aq;
                sA[q * LDA + j] = s * ap + c * aq;
            }
            __syncthreads();
        }
    }

    // ---- ReEig activation on eigenvalues (faithful to reference) ----
    if (t < DIM) {
        const float lam = sA[t * LDA + t];
        float ev = fmaxf(lam, 0.0f);                       // relu
        if (ev <= 0.0f) ev = logf(1.0f + expf(ev)) + 1e-4f; // softplus(ev)+1e-4 guard
        sEv[t] = ev;
    }
    __syncthreads();

    // ---- M = Q * diag(ev), overwrite sA ----
    for (int id = t; id < DIM * DIM; id += 256) {
        const int row = id >> 6, col = id & 63;
        sA[row * LDA + col] = sQ[row * LDA + col] * sEv[col];
    }
    __syncthreads();

    // ---- Out = M @ Q^T via V_WMMA_F32_16X16X4_F32 (fp32, full precision) ----
    // 16 tiles of 16x16; 8 waves -> 2 tiles/wave; 16 k-steps of K=4 each.
    const int wave  = t >> 5;
    const int lane  = t & 31;
    const int row16 = lane & 15;            // M-row (A frag) / N-col (B frag, D col)
    const int khalf = (lane >> 4) * 2;      // K base within a 4-wide k-step
    for (int tile = wave * 2; tile < wave * 2 + 2; ++tile) {
        const int tm = (tile >> 2) * 16;
        const int tn = (tile & 3) * 16;
        v8f acc = {};
        #pragma unroll
        for (int k0 = 0; k0 < DIM; k0 += 4) {
            v2f afrag, bfrag;
            // A fragment: M[tm+row16][k0+khalf .. +1]
            afrag.x = sA[(tm + row16) * LDA + k0 + khalf];
            afrag.y = sA[(tm + row16) * LDA + k0 + khalf + 1];
            // B fragment: Q^T[k][n] = Q[tn+row16][k0+khalf .. +1]
            bfrag.x = sQ[(tn + row16) * LDA + k0 + khalf];
            bfrag.y = sQ[(tn + row16) * LDA + k0 + khalf + 1];
            acc = __builtin_amdgcn_wmma_f32_16x16x4_f32(
                /*neg_a=*/false, afrag, /*neg_b=*/false, bfrag,
                /*c_mod=*/(short)0, acc, /*reuse_a=*/false, /*reuse_b=*/false);
        }
        // D layout: VGPR v -> row tm + v + 8*(lane>>4), col tn + (lane&15)
        const int rbase = tm + ((lane >> 4) * 8);
        #pragma unroll
        for (int v = 0; v < 8; ++v) {
            Ob[(rbase + v) * DIM + tn + row16] = acc[v];
        }
    }
}

extern "C" void kernel_launch(void* const* d_in, const int* in_sizes, int n_in,
                              void* d_out, int out_size, void* d_ws, size_t ws_size,
                              hipStream_t stream) {
    (void)n_in; (void)out_size; (void)d_ws; (void)ws_size;
    const float* X = (const float*)d_in[0];
    float* Out = (float*)d_out;
    const int batch = in_sizes[0] / (DIM * DIM);   // 8192
    eigact_reeig_kernel<<<batch, 256, 0, stream>>>(X, Out);
}